// aten_sparse_mm_59210419142893
// MI455X (gfx1250) — compile-verified
//
#include <hip/hip_runtime.h>

#define BLOCK 256
#define CHUNK 128          // nnz per tile (one async b64 per thread per tile)
#define SPMM_GRID 2048
#define NDIM 128           // dense feature dim (reference: N = 128)

__global__ __launch_bounds__(BLOCK)
void spmm_zero_kernel(float4* __restrict__ out, long long n4) {
    long long i = (long long)blockIdx.x * BLOCK + threadIdx.x;
    if (i < n4) out[i] = make_float4(0.f, 0.f, 0.f, 0.f);
}

__global__ __launch_bounds__(BLOCK)
void spmm_coo_kernel(const long long* __restrict__ rows,
                     const long long* __restrict__ cols,
                     const float*     __restrict__ vals,
                     const float*     __restrict__ dense,
                     float*           __restrict__ out,
                     long long nnz) {
    // Double-buffered LDS staging of the (row,col) stream, interleaved so the
    // consume side reads one b128 per nnz. Filled via CDNA5 async global->LDS DMA.
    __shared__ long long s_rc[2][CHUNK][2];

    const int tid  = threadIdx.x;
    const int lane = tid & 31;
    const int wave = tid >> 5;                    // 0..7 (wave32)
    const long long ntiles = (nnz + CHUNK - 1) / CHUNK;

    // ---- stage one tile: 128 rows + 128 cols, one async b64 per thread ----
    auto stage = [&](long long tile, int buf) {
        const long long base = tile * CHUNK;
        const int j = tid & (CHUNK - 1);
        long long idx = base + j;
        if (idx >= nnz) idx = nnz - 1;            // clamp: harmless valid-memory load
        const long long* src = (tid < CHUNK) ? (rows + idx) : (cols + idx);
        const void* dstp = &s_rc[buf][j][(tid < CHUNK) ? 0 : 1];
        unsigned lds_off = (unsigned)(size_t)dstp;              // low 32 bits = LDS offset
        unsigned long long gaddr = (unsigned long long)(size_t)src;
        // CDNA5 async DMA: memory -> LDS, tracked by ASYNCcnt (one per wave per tile)
        asm volatile("global_load_async_to_lds_b64 %0, %1, off"
                     :: "v"(lds_off), "v"(gaddr)
                     : "memory");
    };

    // ---- consume one tile: batched gathers for MLP, then scale + atomic scatter ----
    auto process = [&](long long tile, int buf) {
        const long long base = tile * CHUNK;
        // prefetch this tile's gathered dense rows into cache
        if (tid < CHUNK) {
            long long c = s_rc[buf][tid][1];
            __builtin_prefetch(dense + (size_t)c * NDIM, 0, 1);
        }
        #pragma unroll
        for (int g = 0; g < 2; ++g) {             // 2 groups of 8 nnz per wave
            const int k0 = wave * (CHUNK / 8) + g * 8;
            long long r[8];
            float     v[8];
            float4    d[8];
            // phase 1: 8 independent b128 gathers in flight (branch-free)
            #pragma unroll
            for (int j = 0; j < 8; ++j) {
                const int k = k0 + j;
                long long e = base + k;
                const long long ec = (e < nnz) ? e : nnz - 1;
                r[j] = s_rc[buf][k][0];           // one ds_load_b128 broadcast pair
                const long long c = s_rc[buf][k][1];
                v[j] = vals[ec];                  // contiguous stream, cache hit
                d[j] = ((const float4*)(dense + (size_t)c * NDIM))[lane]; // global_load_b128
            }
            // phase 2: scale + scatter via native fp32 L2 atomics
            #pragma unroll
            for (int j = 0; j < 8; ++j) {
                const long long e = base + k0 + j;
                if (e < nnz) {
                    float* o = out + (size_t)r[j] * NDIM + lane * 4;
                    unsafeAtomicAdd(o + 0, v[j] * d[j].x);  // global_atomic_add_f32
                    unsafeAtomicAdd(o + 1, v[j] * d[j].y);
                    unsafeAtomicAdd(o + 2, v[j] * d[j].z);
                    unsafeAtomicAdd(o + 3, v[j] * d[j].w);
                }
            }
        }
    };

    long long tile = blockIdx.x;
    if (tile >= ntiles) return;                    // block-uniform exit

    stage(tile, 0);
    int buf = 0;
    for (long long t = tile; t < ntiles; t += gridDim.x) {
        const long long tn = t + gridDim.x;
        const bool have_next = (tn < ntiles);      // block-uniform
        if (have_next) {
            stage(tn, buf ^ 1);
            // keep next tile in flight; wait only for the current tile (in-order)
            asm volatile("s_wait_asynccnt 0x1" ::: "memory");
        } else {
            asm volatile("s_wait_asynccnt 0x0" ::: "memory");
        }
        __syncthreads();                           // staged LDS visible to all 8 waves
        process(t, buf);
        __syncthreads();                           // buffer free before it is re-staged
        buf ^= 1;
    }
}

extern "C" void kernel_launch(void* const* d_in, const int* in_sizes, int n_in,
                              void* d_out, int out_size, void* d_ws, size_t ws_size,
                              hipStream_t stream) {
    // setup_inputs order: indices (2 x NNZ, int64), values (NNZ, f32), dense_mat (K x 128, f32)
    const long long nnz   = (long long)in_sizes[1];
    const long long* rows = (const long long*)d_in[0];
    const long long* cols = rows + nnz;
    const float* vals     = (const float*)d_in[1];
    const float* dense    = (const float*)d_in[2];
    float* out            = (float*)d_out;

    // zero-init output (harness poisons d_out with 0xAA)
    const long long n4 = (long long)out_size / 4;
    const unsigned zgrid = (unsigned)((n4 + BLOCK - 1) / BLOCK);
    spmm_zero_kernel<<<zgrid, BLOCK, 0, stream>>>((float4*)out, n4);

    spmm_coo_kernel<<<SPMM_GRID, BLOCK, 0, stream>>>(rows, cols, vals, dense, out, nnz);
}